// RiskGCN_81406810129168
// MI455X (gfx1250) — compile-verified
//
#include <hip/hip_runtime.h>
#include <hip/hip_bf16.h>

typedef float v2f __attribute__((ext_vector_type(2)));
typedef float v8f __attribute__((ext_vector_type(8)));

// ---------------- degree / normalization ----------------

__global__ void k_set_deg(float* __restrict__ deg, int n) {
  int i = blockIdx.x * blockDim.x + threadIdx.x;
  if (i < n) deg[i] = 1.0f;  // self-loop contributes 1
}

__global__ void k_deg_edges(const long long* __restrict__ dst,
                            float* __restrict__ deg, int e) {
  int i = blockIdx.x * blockDim.x + threadIdx.x;
  if (i < e) atomicAdd(&deg[(int)dst[i]], 1.0f);
}

__global__ void k_dinv(float* __restrict__ deg, int n) {
  int i = blockIdx.x * blockDim.x + threadIdx.x;
  if (i < n) {
    float d = deg[i];
    deg[i] = (d > 0.0f) ? rsqrtf(d) : 0.0f;
  }
}

// ---------------- fp32 WMMA GEMM: H[M x CN] = X[M x K] * W[K x CN] ---------
// One wave computes one 16x16 output tile; K consumed in steps of 4 via
// V_WMMA_F32_16X16X4_F32. M must be a multiple of 16 (50000 = 3125*16).

template <int K, int CN>
__global__ void __launch_bounds__(256)
k_gemm_wmma(const float* __restrict__ X, const float* __restrict__ W,
            float* __restrict__ H, int M) {
  const int lane = threadIdx.x & 31;
  const int wave = blockIdx.x * (blockDim.x >> 5) + (threadIdx.x >> 5);
  constexpr int NT = CN / 16;            // N-tiles per M-tile row
  const int mtile = wave / NT;
  const int ntile = wave % NT;
  if (mtile * 16 >= M) return;           // wave-uniform exit; EXEC stays full

  const int row   = mtile * 16 + (lane & 15);  // A row for this lane
  const int ncol  = ntile * 16 + (lane & 15);  // B/D column for this lane
  const int khalf = (lane >> 4) * 2;           // 0 for lanes 0-15, 2 for 16-31

  v8f acc = {};
  const float* __restrict__ xr = X + (size_t)row * K;
#pragma unroll 4
  for (int k0 = 0; k0 < K; k0 += 4) {
    const int ka = k0 + khalf;
    v2f a, b;
    a.x = xr[ka];
    a.y = xr[ka + 1];
    b.x = W[(size_t)ka * CN + ncol];
    b.y = W[(size_t)(ka + 1) * CN + ncol];
    acc = __builtin_amdgcn_wmma_f32_16x16x4_f32(
        /*neg_a=*/false, a, /*neg_b=*/false, b,
        /*c_mod=*/(short)0, acc, /*reuse_a=*/false, /*reuse_b=*/false);
  }

  const int mbase = mtile * 16 + (lane >> 4) * 8;  // D: m = v + 8*(lane/16)
#pragma unroll
  for (int v = 0; v < 8; ++v)
    H[(size_t)(mbase + v) * CN + ncol] = acc[v];
}

// ---------------- self-loop + bias init:  out = h*dinv^2 + bias -----------

template <int C>
__global__ void k_self_init(const float* __restrict__ h,
                            const float* __restrict__ dinv,
                            const float* __restrict__ bias,
                            float* __restrict__ out, int n) {
  int idx = blockIdx.x * blockDim.x + threadIdx.x;
  if (idx >= n * C) return;
  int node = idx / C;
  int c = idx - node * C;
  float di = dinv[node];
  out[idx] = fmaf(h[idx], di * di, bias[c]);
}

// ---------------- edge scatter: one wave per edge -------------------------
// Lane l handles channels [l*PER, l*PER+PER): coalesced gather, f32 atomics.

template <int C>
__global__ void __launch_bounds__(256)
k_scatter(const long long* __restrict__ src, const long long* __restrict__ dst,
          const float* __restrict__ dinv, const float* __restrict__ h,
          float* __restrict__ out, int e) {
  const int wave = blockIdx.x * (blockDim.x >> 5) + (threadIdx.x >> 5);
  const int lane = threadIdx.x & 31;
  if (wave >= e) return;
  const int s = (int)src[wave];
  const int d = (int)dst[wave];
  const float w = dinv[s] * dinv[d];
  constexpr int PER = C / 32;
  const float* __restrict__ hp = h + (size_t)s * C + lane * PER;
  float* __restrict__ op = out + (size_t)d * C + lane * PER;
#pragma unroll
  for (int i = 0; i < PER; ++i) atomicAdd(op + i, hp[i] * w);
}

__global__ void k_relu(float* __restrict__ a, int n) {
  int i = blockIdx.x * blockDim.x + threadIdx.x;
  if (i < n) a[i] = fmaxf(a[i], 0.0f);
}

// ---------------- driver --------------------------------------------------

extern "C" void kernel_launch(void* const* d_in, const int* in_sizes, int n_in,
                              void* d_out, int out_size, void* d_ws,
                              size_t ws_size, hipStream_t stream) {
  const float* x        = (const float*)d_in[0];
  const long long* eidx = (const long long*)d_in[1];   // [2, E] int64
  const float* W1       = (const float*)d_in[2];
  const float* b1       = (const float*)d_in[3];
  const float* W2       = (const float*)d_in[4];
  const float* b2       = (const float*)d_in[5];
  float* out            = (float*)d_out;

  constexpr int IN_C = 128, HID_C = 128, OUT_C = 64;
  const int N = in_sizes[0] / IN_C;   // 50000
  const int E = in_sizes[1] / 2;      // 800000
  const long long* esrc = eidx;
  const long long* edst = eidx + E;

  float* ws   = (float*)d_ws;
  float* dinv = ws;                       // N   (degree, then rsqrt in place)
  float* h    = dinv + N;                 // N*128 (layer-1 pre-agg, reused for layer-2)
  float* a1   = h + (size_t)N * HID_C;    // N*128 (layer-1 post-agg)

  const int TPB = 256;

  // deg -> dinv
  k_set_deg<<<(N + TPB - 1) / TPB, TPB, 0, stream>>>(dinv, N);
  k_deg_edges<<<(E + TPB - 1) / TPB, TPB, 0, stream>>>(edst, dinv, E);
  k_dinv<<<(N + TPB - 1) / TPB, TPB, 0, stream>>>(dinv, N);

  // layer 1: h = x @ W1  (WMMA fp32), then aggregate
  {
    const int waves = (N / 16) * (HID_C / 16);
    k_gemm_wmma<IN_C, HID_C><<<(waves + 7) / 8, TPB, 0, stream>>>(x, W1, h, N);
    k_self_init<HID_C><<<((size_t)N * HID_C + TPB - 1) / TPB, TPB, 0, stream>>>(
        h, dinv, b1, a1, N);
    k_scatter<HID_C><<<(E + 7) / 8, TPB, 0, stream>>>(esrc, edst, dinv, h, a1, E);
    k_relu<<<((size_t)N * HID_C + TPB - 1) / TPB, TPB, 0, stream>>>(a1, N * HID_C);
  }

  // layer 2: h2 = relu(a1) @ W2 (reuse h buffer), aggregate into d_out
  {
    const int waves = (N / 16) * (OUT_C / 16);
    k_gemm_wmma<HID_C, OUT_C><<<(waves + 7) / 8, TPB, 0, stream>>>(a1, W2, h, N);
    k_self_init<OUT_C><<<((size_t)N * OUT_C + TPB - 1) / TPB, TPB, 0, stream>>>(
        h, dinv, b2, out, N);
    k_scatter<OUT_C><<<(E + 7) / 8, TPB, 0, stream>>>(esrc, edst, dinv, h, out, E);
  }
}